// AxialAttentionWithoutPosition3D_51316269252943
// MI455X (gfx1250) — compile-verified
//
#include <hip/hip_runtime.h>
#include <hip/hip_bf16.h>

// ---------------------------------------------------------------------------
// AxialAttentionWithoutPosition3D for MI455X (gfx1250), fp32 throughout.
// All three GEMMs run on V_WMMA_F32_16X16X4_F32 (exact fp32 semantics, RNE).
// Multi-kernel pipeline because the three BatchNorms need global statistics.
// ---------------------------------------------------------------------------

typedef __attribute__((ext_vector_type(2))) float v2f;
typedef __attribute__((ext_vector_type(8))) float v8f;

__device__ __forceinline__ v8f wmma_f32(v2f a, v2f b, v8f c) {
    // 8 args: (neg_a, A, neg_b, B, c_mod, C, reuse_a, reuse_b)
    return __builtin_amdgcn_wmma_f32_16x16x4_f32(
        false, a, false, b, (short)0, c, false, false);
}

// Problem constants
#define BATCH   3136          // 56*56 (A*D)
#define HDIM    56
#define CIN     128
#define OQKV    256           // 2*OUT_PLANES
#define OUTP    128
#define NGRP    8
#define ELEMS_PER_CHAN 175616 // BATCH*HDIM
#define QKV_ROW (OQKV*HDIM)   // 14336
#define SV_ROW  (OUTP*HDIM)   // 7168

// Workspace layout (floats)
#define QKV_OFF     0
#define SV_OFF      44957696              // BATCH*QKV_ROW
#define BNQKV_OFF   (SV_OFF + 22478848)   // + BATCH*SV_ROW
#define QKSTATS_OFF (BNQKV_OFF + 512)
#define BNQK_OFF    (QKSTATS_OFF + 16)
#define BNOUT_OFF   (BNQK_OFF + 16)

// ---------------------------------------------------------------------------
__global__ void zero_kernel(float* __restrict__ p, int n) {
    int i = blockIdx.x * blockDim.x + threadIdx.x;
    if (i < n) p[i] = 0.f;
}

// ---------------------------------------------------------------------------
// K1: qkv[b, o, h] = sum_c W[o,c] * x[c, a, h, d]   (b = a*56 + d)
// Block: (32,16) = 16 waves; wave y owns O-tile y. X tile staged in LDS once.
// Grid: (BATCH, 4 h-tiles).
__global__ __launch_bounds__(512) void qkv_gemm_kernel(
    const float* __restrict__ x, const float* __restrict__ w,
    float* __restrict__ qkv)
{
    __shared__ float Xs[CIN][16];
    const int b    = blockIdx.x;
    const int ht   = blockIdx.y;
    const int lane = threadIdx.x;
    const int ot   = threadIdx.y;
    const int tid  = ot * 32 + lane;
    const int a = b / HDIM, d = b % HDIM;
    const float* xb = x + a * 3136 + d;   // + c*175616 + h*56

    for (int idx = tid; idx < CIN * 16; idx += 512) {
        int c = idx >> 4, hl = idx & 15;
        int h = ht * 16 + hl;
        Xs[c][hl] = (h < HDIM) ? xb[(size_t)c * ELEMS_PER_CHAN + h * HDIM] : 0.f;
    }
    __syncthreads();

    const int half = lane >> 4, l = lane & 15;
    const int m = ot * 16 + l;            // output channel row for A
    v8f acc = {};
    for (int k = 0; k < CIN; k += 4) {
        int ka = k + 2 * half;
        v2f av, bv;
        av.x = w[m * CIN + ka];
        av.y = w[m * CIN + ka + 1];
        bv.x = Xs[ka][l];
        bv.y = Xs[ka + 1][l];
        acc = wmma_f32(av, bv, acc);
    }

    float* qb = qkv + (size_t)b * QKV_ROW;
    int h = ht * 16 + l;
    if (h < HDIM) {
#pragma unroll
        for (int vi = 0; vi < 8; ++vi) {
            int row = ot * 16 + 8 * half + vi;   // D: VGPR vi -> M = vi + 8*half
            qb[row * HDIM + h] = acc[vi];
        }
    }
}

// ---------------------------------------------------------------------------
// K2/K6: deterministic per-channel mean/var -> fused BN scale/shift.
// One 256-thread block per channel.
__global__ __launch_bounds__(256) void chan_stats_kernel(
    const float* __restrict__ data, int nchan,
    const float* __restrict__ gamma, const float* __restrict__ beta,
    float* __restrict__ bn)
{
    __shared__ float s1[256], s2[256];
    const int ch = blockIdx.x, tid = threadIdx.x;
    const size_t rowstride = (size_t)nchan * HDIM;
    float sum = 0.f, sq = 0.f;
    for (int idx = tid; idx < ELEMS_PER_CHAN; idx += 256) {
        int bb = idx / HDIM, h = idx - bb * HDIM;
        float v = data[(size_t)bb * rowstride + (size_t)ch * HDIM + h];
        sum += v; sq += v * v;
    }
    s1[tid] = sum; s2[tid] = sq;
    __syncthreads();
    for (int s = 128; s > 0; s >>= 1) {
        if (tid < s) { s1[tid] += s1[tid + s]; s2[tid] += s2[tid + s]; }
        __syncthreads();
    }
    if (tid == 0) {
        const float cnt = (float)ELEMS_PER_CHAN;
        float mean = s1[0] / cnt;
        float var  = s2[0] / cnt - mean * mean;
        float sc   = gamma[ch] * rsqrtf(var + 1e-5f);
        bn[ch]         = sc;
        bn[nchan + ch] = beta[ch] - mean * sc;
    }
}

// ---------------------------------------------------------------------------
// K3: per (b,g): qk = q^T k (K=8) via WMMA; accumulate group sum/sumsq.
// One wave per (b,g).
__global__ __launch_bounds__(32) void qk_stats_kernel(
    const float* __restrict__ qkv, const float* __restrict__ bnqkv,
    float* __restrict__ qkstats)
{
    __shared__ float qL[8][65];
    __shared__ float kL[8][65];
    const int b = blockIdx.x, g = blockIdx.y;
    const int lane = threadIdx.x;
    const int half = lane >> 4, l = lane & 15;
    const float* base = qkv + (size_t)b * QKV_ROW + g * 32 * HDIM;

    for (int idx = lane; idx < 16 * 64; idx += 32) {
        int c = idx >> 6, h = idx & 63;
        float t = 0.f;
        if (h < HDIM)
            t = base[c * HDIM + h] * bnqkv[g * 32 + c] + bnqkv[OQKV + g * 32 + c];
        if (c < 8) qL[c][h] = t; else kL[c - 8][h] = t;
    }
    __syncthreads();

    float sum = 0.f, sq = 0.f;
    for (int it = 0; it < 4; ++it)
        for (int jt = 0; jt < 4; ++jt) {
            v8f acc = {};
#pragma unroll
            for (int kk = 0; kk < 8; kk += 4) {
                v2f av, bv;
                av.x = qL[kk + 2 * half][it * 16 + l];
                av.y = qL[kk + 2 * half + 1][it * 16 + l];
                bv.x = kL[kk + 2 * half][jt * 16 + l];
                bv.y = kL[kk + 2 * half + 1][jt * 16 + l];
                acc = wmma_f32(av, bv, acc);
            }
            int j = jt * 16 + l;
#pragma unroll
            for (int vi = 0; vi < 8; ++vi) {
                int i = it * 16 + 8 * half + vi;
                float v = acc[vi];
                if (i < HDIM && j < HDIM) { sum += v; sq += v * v; }
            }
        }
    for (int mask = 1; mask < 32; mask <<= 1) {
        sum += __shfl_xor(sum, mask, 32);
        sq  += __shfl_xor(sq,  mask, 32);
    }
    if (lane == 0) {
        atomicAdd(&qkstats[g], sum);
        atomicAdd(&qkstats[NGRP + g], sq);
    }
}

__global__ void qk_finalize_kernel(const float* __restrict__ stats,
                                   const float* __restrict__ gamma,
                                   const float* __restrict__ beta,
                                   float* __restrict__ bn)
{
    int g = threadIdx.x;
    if (g < NGRP) {
        const float cnt = 9834496.f;   // BATCH * 56 * 56
        float mean = stats[g] / cnt;
        float var  = stats[NGRP + g] / cnt - mean * mean;
        float sc   = gamma[g] * rsqrtf(var + 1e-5f);
        bn[g]        = sc;
        bn[NGRP + g] = beta[g] - mean * sc;
    }
}

// ---------------------------------------------------------------------------
// K5: per (b,g): recompute qk (WMMA), group-BN + softmax in LDS,
// sv = V * sim^T (WMMA), write sv. One wave per (b,g).
__global__ __launch_bounds__(32) void attn_kernel(
    const float* __restrict__ qkv, const float* __restrict__ bnqkv,
    const float* __restrict__ bnqk, float* __restrict__ sv)
{
    __shared__ float qL[8][65];
    __shared__ float kL[8][65];
    __shared__ float vL[16][65];
    __shared__ float simL[64][65];
    const int b = blockIdx.x, g = blockIdx.y;
    const int lane = threadIdx.x;
    const int half = lane >> 4, l = lane & 15;
    const float* base = qkv + (size_t)b * QKV_ROW + g * 32 * HDIM;

    for (int idx = lane; idx < 32 * 64; idx += 32) {
        int c = idx >> 6, h = idx & 63;
        float t = 0.f;
        if (h < HDIM)
            t = base[c * HDIM + h] * bnqkv[g * 32 + c] + bnqkv[OQKV + g * 32 + c];
        if (c < 8)       qL[c][h]      = t;
        else if (c < 16) kL[c - 8][h]  = t;
        else             vL[c - 16][h] = t;
    }
    __syncthreads();

    // qk tiles -> simL (raw, BN applied in softmax pass)
    for (int it = 0; it < 4; ++it)
        for (int jt = 0; jt < 4; ++jt) {
            v8f acc = {};
#pragma unroll
            for (int kk = 0; kk < 8; kk += 4) {
                v2f av, bv;
                av.x = qL[kk + 2 * half][it * 16 + l];
                av.y = qL[kk + 2 * half + 1][it * 16 + l];
                bv.x = kL[kk + 2 * half][jt * 16 + l];
                bv.y = kL[kk + 2 * half + 1][jt * 16 + l];
                acc = wmma_f32(av, bv, acc);
            }
#pragma unroll
            for (int vi = 0; vi < 8; ++vi)
                simL[it * 16 + 8 * half + vi][jt * 16 + l] = acc[vi];
        }
    __syncthreads();

    // group BN + softmax over j; zero the padding so it cannot contribute
    const float gsc = bnqk[g], gsh = bnqk[NGRP + g];
    for (int r = lane; r < 64; r += 32) {
        if (r < HDIM) {
            float mx = -3.0e38f;
            for (int j = 0; j < HDIM; ++j) {
                float t = simL[r][j] * gsc + gsh;
                simL[r][j] = t;
                mx = fmaxf(mx, t);
            }
            float s = 0.f;
            for (int j = 0; j < HDIM; ++j) {
                float e = __expf(simL[r][j] - mx);
                simL[r][j] = e;
                s += e;
            }
            float inv = 1.f / s;
            for (int j = 0; j < HDIM; ++j) simL[r][j] *= inv;
            for (int j = HDIM; j < 64; ++j) simL[r][j] = 0.f;
        } else {
            for (int j = 0; j < 64; ++j) simL[r][j] = 0.f;
        }
    }
    __syncthreads();

    // sv[c,i] = sum_j vL[c][j] * simL[i][j]  -> A = V (16xK), B = sim^T (Kx16)
    float* svb = sv + (size_t)b * SV_ROW + g * 16 * HDIM;
    for (int nt = 0; nt < 4; ++nt) {
        v8f acc = {};
        for (int kk = 0; kk < 64; kk += 4) {
            v2f av, bv;
            av.x = vL[l][kk + 2 * half];
            av.y = vL[l][kk + 2 * half + 1];
            bv.x = simL[nt * 16 + l][kk + 2 * half];
            bv.y = simL[nt * 16 + l][kk + 2 * half + 1];
            acc = wmma_f32(av, bv, acc);
        }
        int i = nt * 16 + l;
        if (i < HDIM) {
#pragma unroll
            for (int vi = 0; vi < 8; ++vi) {
                int c = 8 * half + vi;
                svb[c * HDIM + i] = acc[vi];
            }
        }
    }
}

// ---------------------------------------------------------------------------
// K8: final BN + transposed store: out[oc, a, h, d] = bn(sv[a*56+d, oc, h])
__global__ __launch_bounds__(256) void writeout_kernel(
    const float* __restrict__ sv, const float* __restrict__ bn,
    float* __restrict__ out)
{
    int tid = blockIdx.x * 256 + threadIdx.x;
    if (tid >= OUTP * ELEMS_PER_CHAN) return;
    int d = tid % HDIM; int t = tid / HDIM;
    int h = t % HDIM;   t /= HDIM;
    int a = t % HDIM;   int oc = t / HDIM;
    float v = sv[(size_t)(a * HDIM + d) * SV_ROW + oc * HDIM + h];
    out[tid] = v * bn[oc] + bn[OUTP + oc];
}

// ---------------------------------------------------------------------------
extern "C" void kernel_launch(void* const* d_in, const int* in_sizes, int n_in,
                              void* d_out, int out_size, void* d_ws, size_t ws_size,
                              hipStream_t stream) {
    const float* x     = (const float*)d_in[0];
    const float* w     = (const float*)d_in[1];
    const float* g_qkv = (const float*)d_in[2];
    const float* b_qkv = (const float*)d_in[3];
    const float* g_sim = (const float*)d_in[4];
    const float* b_sim = (const float*)d_in[5];
    const float* g_out = (const float*)d_in[6];
    const float* b_out = (const float*)d_in[7];
    float* out = (float*)d_out;
    float* ws  = (float*)d_ws;

    float* qkv     = ws + QKV_OFF;
    float* sv      = ws + SV_OFF;
    float* bnqkv   = ws + BNQKV_OFF;
    float* qkstats = ws + QKSTATS_OFF;
    float* bnqk    = ws + BNQK_OFF;
    float* bnout   = ws + BNOUT_OFF;

    zero_kernel<<<1, 32, 0, stream>>>(qkstats, 16);
    qkv_gemm_kernel<<<dim3(BATCH, 4), dim3(32, 16), 0, stream>>>(x, w, qkv);
    chan_stats_kernel<<<OQKV, 256, 0, stream>>>(qkv, OQKV, g_qkv, b_qkv, bnqkv);
    qk_stats_kernel<<<dim3(BATCH, NGRP), 32, 0, stream>>>(qkv, bnqkv, qkstats);
    qk_finalize_kernel<<<1, 8, 0, stream>>>(qkstats, g_sim, b_sim, bnqk);
    attn_kernel<<<dim3(BATCH, NGRP), 32, 0, stream>>>(qkv, bnqkv, bnqk, sv);
    chan_stats_kernel<<<OUTP, 256, 0, stream>>>(sv, OUTP, g_out, b_out, bnout);
    writeout_kernel<<<(OUTP * ELEMS_PER_CHAN + 255) / 256, 256, 0, stream>>>(sv, bnout, out);
}